// Simpler_GCN_Conv_4492535792525
// MI455X (gfx1250) — compile-verified
//
#include <hip/hip_runtime.h>
#include <hip/hip_bf16.h>

typedef float v2f __attribute__((ext_vector_type(2)));
typedef float v8f __attribute__((ext_vector_type(8)));

#define LEAKY(v) ((v) > 0.0f ? (v) : 0.1f * (v))

// ---------------------------------------------------------------------------
// init: deg (3N floats) = 1.0 (self loop), acc (15N floats) = 0
// ---------------------------------------------------------------------------
__global__ void gcn_init_kernel(float* __restrict__ deg, float* __restrict__ acc, int n3, int n15) {
    int i = blockIdx.x * blockDim.x + threadIdx.x;
    if (i < n3)  deg[i] = 1.0f;
    if (i < n15) acc[i] = 0.0f;
}

// ---------------------------------------------------------------------------
// degree: atomicAdd 1.0 per edge into deg[dst].  dst stream is read-once ->
// non-temporal so it does not evict the L2-resident deg/h/acc working set.
// ---------------------------------------------------------------------------
__global__ void gcn_degree_kernel(const int* __restrict__ dst, float* __restrict__ deg, int E) {
    int e = blockIdx.x * blockDim.x + threadIdx.x;
    if (e >= E) return;
    int d = __builtin_nontemporal_load(dst + e);
    __hip_atomic_fetch_add(&deg[d], 1.0f, __ATOMIC_RELAXED, __HIP_MEMORY_SCOPE_AGENT);
}

// ---------------------------------------------------------------------------
// rsqrt in place: deg -> dinv   (deg >= 1 always due to self loops)
// ---------------------------------------------------------------------------
__global__ void gcn_rsqrt_kernel(float* __restrict__ d, int n3) {
    int i = blockIdx.x * blockDim.x + threadIdx.x;
    if (i < n3) d[i] = rsqrtf(d[i]);
}

// ---------------------------------------------------------------------------
// h = x @ [W_p | W_s | W_v]  via V_WMMA_F32_16X16X4_F32.
// Block = 256 threads = 8 waves = 128 rows of x.
// Stage: ldsX[128][28] (K-padded x tile, coalesced NT fill) and
//        ldsW[28][16]  (combined weights, col 15 = 0).
// Inner loop: branch-free ds_load_b64 (A) + ds_load_2addr (B) + 7 chained WMMAs.
// ---------------------------------------------------------------------------
__global__ void gcn_gemm_wmma_kernel(const float* __restrict__ x,
                                     const float* __restrict__ W_p,
                                     const float* __restrict__ W_s,
                                     const float* __restrict__ W_v,
                                     float* __restrict__ h, int N) {
    __shared__ __align__(16) float ldsX[128 * 28];
    __shared__ __align__(16) float ldsW[28 * 16];

    const int rowblock = blockIdx.x * 128;

    // ---- stage weights: ldsW[k*16 + n], zero padded (k>=25 or n==15) ----
    for (int idx = threadIdx.x; idx < 28 * 16; idx += blockDim.x) {
        int k = idx >> 4;
        int n = idx & 15;
        float w = 0.0f;
        if (k < 25 && n < 15) {
            int g = n / 5;
            int c = n - g * 5;
            const float* Wg = (g == 0) ? W_p : (g == 1) ? W_s : W_v;
            w = Wg[k * 5 + c];
        }
        ldsW[idx] = w;
    }

    // ---- stage x tile: contiguous 128*25 floats, fully coalesced, NT ----
    for (int idx = threadIdx.x; idx < 128 * 25; idx += blockDim.x) {
        int r = idx / 25;
        int k = idx - r * 25;
        int grow = rowblock + r;
        float v = (grow < N) ? __builtin_nontemporal_load(x + (size_t)grow * 25 + k) : 0.0f;
        ldsX[r * 28 + k] = v;
    }
    // zero the K-pad columns 25..27
    for (int idx = threadIdx.x; idx < 128 * 3; idx += blockDim.x) {
        int r = idx / 3;
        int k = 25 + (idx - r * 3);
        ldsX[r * 28 + k] = 0.0f;
    }
    __syncthreads();

    const int lane = threadIdx.x & 31;
    const int wave = threadIdx.x >> 5;
    const int rowbase = rowblock + wave * 16;

    const int rl   = wave * 16 + (lane & 15);   // local A row
    const int kofs = (lane >> 4) * 2;           // lanes 0-15: K=0,1 ; 16-31: K=2,3
    const int colN = lane & 15;                 // B/D column

    v8f acc = {};
    #pragma unroll
    for (int kc = 0; kc < 28; kc += 4) {
        int k = kc + kofs;                      // even, <= 26 -> 8B-aligned b64 load
        v2f a = *(const v2f*)&ldsX[rl * 28 + k];
        v2f b;
        b.x = ldsW[k * 16 + colN];
        b.y = ldsW[(k + 1) * 16 + colN];
        // 8 args: (neg_a, A, neg_b, B, c_mod, C, reuse_a, reuse_b)
        acc = __builtin_amdgcn_wmma_f32_16x16x4_f32(false, a, false, b,
                                                    (short)0, acc, false, false);
    }

    // D layout: VGPR r -> lanes 0-15: M=r, N=lane ; lanes 16-31: M=8+r, N=lane-16
    const size_t N5 = (size_t)N * 5;
    const int radd = (lane >> 4) ? 8 : 0;
    #pragma unroll
    for (int r = 0; r < 8; ++r) {
        int row = rowbase + r + radd;
        if (row < N && colN < 15) {
            int g = colN / 5;
            int c = colN - g * 5;
            h[(size_t)g * N5 + (size_t)row * 5 + c] = acc[r];
        }
    }
}

// ---------------------------------------------------------------------------
// edge scatter: acc[dst] += h[src] * dinv[src]*dinv[dst]   (5 FP32 atomics).
// Index streams NT; h/dinv/acc stay L2-resident.
// ---------------------------------------------------------------------------
__global__ void gcn_scatter_kernel(const int* __restrict__ src,
                                   const int* __restrict__ dst,
                                   const float* __restrict__ dinv,
                                   const float* __restrict__ h,
                                   float* __restrict__ acc, int E) {
    int e = blockIdx.x * blockDim.x + threadIdx.x;
    if (e >= E) return;
    int s = __builtin_nontemporal_load(src + e);
    int d = __builtin_nontemporal_load(dst + e);
    float w = dinv[s] * dinv[d];
    const float* hs = h + (size_t)s * 5;
    float* ad = acc + (size_t)d * 5;
    #pragma unroll
    for (int c = 0; c < 5; ++c) {
        __hip_atomic_fetch_add(ad + c, hs[c] * w, __ATOMIC_RELAXED, __HIP_MEMORY_SCOPE_AGENT);
    }
}

// ---------------------------------------------------------------------------
// finalize: per node add self-loop term + bias, leaky, then the full MLP.
// ---------------------------------------------------------------------------
__global__ void gcn_finalize_kernel(const float* __restrict__ dinv,   // [3N]
                                    const float* __restrict__ h,      // [3][N,5]
                                    const float* __restrict__ acc,    // [3][N,5]
                                    const float* __restrict__ b_p,
                                    const float* __restrict__ b_s,
                                    const float* __restrict__ b_v,
                                    const float* __restrict__ pW1,    // [15,10]
                                    const float* __restrict__ pb1,    // [10]
                                    const float* __restrict__ pW2,    // [10,5]
                                    const float* __restrict__ pb2,    // [5]
                                    const float* __restrict__ cW1,    // [5,5]
                                    const float* __restrict__ cb1,    // [5]
                                    const float* __restrict__ cW2,    // [5,2]
                                    const float* __restrict__ cb2,    // [2]
                                    float* __restrict__ out, int N) {
    int i = blockIdx.x * blockDim.x + threadIdx.x;
    if (i >= N) return;
    const size_t N5 = (size_t)N * 5;

    float x15[15];
    #pragma unroll
    for (int g = 0; g < 3; ++g) {
        const float* bg = (g == 0) ? b_p : (g == 1) ? b_s : b_v;
        float di = dinv[(size_t)g * N + i];
        float dii = di * di;
        const float* hg = h + (size_t)g * N5 + (size_t)i * 5;
        const float* ag = acc + (size_t)g * N5 + (size_t)i * 5;
        #pragma unroll
        for (int c = 0; c < 5; ++c) {
            float v = ag[c] + hg[c] * dii + bg[c];
            x15[g * 5 + c] = LEAKY(v);
        }
    }

    float t10[10];
    #pragma unroll
    for (int j = 0; j < 10; ++j) {
        float s = pb1[j];
        #pragma unroll
        for (int k = 0; k < 15; ++k) s += x15[k] * pW1[k * 10 + j];
        t10[j] = LEAKY(s);
    }
    float t5[5];
    #pragma unroll
    for (int j = 0; j < 5; ++j) {
        float s = pb2[j];
        #pragma unroll
        for (int k = 0; k < 10; ++k) s += t10[k] * pW2[k * 5 + j];
        t5[j] = s;
    }
    float u5[5];
    #pragma unroll
    for (int j = 0; j < 5; ++j) {
        float s = cb1[j];
        #pragma unroll
        for (int k = 0; k < 5; ++k) s += t5[k] * cW1[k * 5 + j];
        u5[j] = LEAKY(s);
    }
    #pragma unroll
    for (int j = 0; j < 2; ++j) {
        float s = cb2[j];
        #pragma unroll
        for (int k = 0; k < 5; ++k) s += u5[k] * cW2[k * 2 + j];
        out[(size_t)i * 2 + j] = s;
    }
}

// ---------------------------------------------------------------------------
extern "C" void kernel_launch(void* const* d_in, const int* in_sizes, int n_in,
                              void* d_out, int out_size, void* d_ws, size_t ws_size,
                              hipStream_t stream) {
    const float* x    = (const float*)d_in[0];
    const int*   ei_p = (const int*)d_in[1];
    const int*   ei_s = (const int*)d_in[2];
    const int*   ei_v = (const int*)d_in[3];
    const float* W_p  = (const float*)d_in[4];
    const float* b_p  = (const float*)d_in[5];
    const float* W_s  = (const float*)d_in[6];
    const float* b_s  = (const float*)d_in[7];
    const float* W_v  = (const float*)d_in[8];
    const float* b_v  = (const float*)d_in[9];
    const float* pW1  = (const float*)d_in[10];
    const float* pb1  = (const float*)d_in[11];
    const float* pW2  = (const float*)d_in[12];
    const float* pb2  = (const float*)d_in[13];
    const float* cW1  = (const float*)d_in[14];
    const float* cb1  = (const float*)d_in[15];
    const float* cW2  = (const float*)d_in[16];
    const float* cb2  = (const float*)d_in[17];
    float* out = (float*)d_out;

    const int N = in_sizes[0] / 25;
    const int E = in_sizes[1] / 2;

    // Workspace layout (floats): dinv[3N] | h[3*N*5] | acc[3*N*5]  = 33N
    float* dinv = (float*)d_ws;
    float* h    = dinv + (size_t)3 * N;
    float* accw = h + (size_t)15 * N;

    const int BS = 256;
    const int n3  = 3 * N;
    const int n15 = 15 * N;

    gcn_init_kernel<<<(n15 + BS - 1) / BS, BS, 0, stream>>>(dinv, accw, n3, n15);

    const int* srcs[3] = { ei_p, ei_s, ei_v };
    const int* dsts[3] = { ei_p + E, ei_s + E, ei_v + E };

    const int egrid = (E + BS - 1) / BS;
    for (int g = 0; g < 3; ++g) {
        gcn_degree_kernel<<<egrid, BS, 0, stream>>>(dsts[g], dinv + (size_t)g * N, E);
    }

    gcn_rsqrt_kernel<<<(n3 + BS - 1) / BS, BS, 0, stream>>>(dinv, n3);

    // 8 waves/block * 16 rows/wave = 128 rows per block
    gcn_gemm_wmma_kernel<<<(N + 127) / 128, BS, 0, stream>>>(x, W_p, W_s, W_v, h, N);

    for (int g = 0; g < 3; ++g) {
        gcn_scatter_kernel<<<egrid, BS, 0, stream>>>(
            srcs[g], dsts[g], dinv + (size_t)g * N,
            h + (size_t)g * N * 5, accw + (size_t)g * N * 5, E);
    }

    gcn_finalize_kernel<<<(N + BS - 1) / BS, BS, 0, stream>>>(
        dinv, h, accw, b_p, b_s, b_v,
        pW1, pb1, pW2, pb2, cW1, cb1, cW2, cb2, out, N);
}